// RandomShiftsAug_9646496546902
// MI455X (gfx1250) — compile-verified
//
#include <hip/hip_runtime.h>

#define PAD 4
#define H 128
#define IMG_ELEMS (H * H)        // 16384 per (n,c) image
#define THREADS 256              // 8 waves (wave32)
#define ROWS_PER_HALF 64         // 64 rows * 512B = 32 KB LDS staging

// Low 32 bits of a generic pointer into LDS are the LDS byte offset
// (aperture lives in addr[63:32]; see ISA §10.2).
__device__ __forceinline__ unsigned lds_byte_off(const void* p) {
    return (unsigned)(size_t)p;
}

__device__ __forceinline__ int clamp127(int v) {
    return v < 0 ? 0 : (v > H - 1 ? H - 1 : v);
}

__global__ void __launch_bounds__(THREADS)
RandomShiftsAug_kernel(const float* __restrict__ x,
                       const int* __restrict__ shift,
                       float* __restrict__ out) {
    // Row-major staging: row r occupies bytes [r*512, r*512+511]
    __shared__ float smem[ROWS_PER_HALF * H];   // 32 KB

    const int img  = blockIdx.x;                // img = n*C + c
    const int n    = img / 9;
    const int t    = threadIdx.x;
    const int wave = t >> 5;
    const int lane = t & 31;

    // shift[n,0] shifts x (columns), shift[n,1] shifts y (rows); both in [0,8].
    // Integer shift => the bilinear grid_sample degenerates to an exact
    // clamped gather: out[i,j] = x[clamp(i+sy-4), clamp(j+sx-4)].
    const int sx = shift[2 * n + 0] - PAD;      // [-4, +4]
    const int sy = shift[2 * n + 1] - PAD;

    const float* __restrict__ src = x   + (size_t)img * IMG_ELEMS;
    float*       __restrict__ dst = out + (size_t)img * IMG_ELEMS;
    const unsigned lbase = lds_byte_off(&smem[0]);

    for (int half = 0; half < 2; ++half) {
        const int i0 = half * ROWS_PER_HALF;

        // ---- async loads: one aligned 512B row per instruction ----
        // wave w loads relative rows r = w*8 .. w*8+7; lane supplies 16B.
        for (int k = 0; k < 8; ++k) {
            const int r  = wave * 8 + k;                   // 0..63
            const int yc = clamp127(i0 + r + sy);          // row clamp (edge pad)
            const float* g  = src + yc * H + lane * 4;     // 16B-aligned
            const unsigned l = lbase + (unsigned)(r * (H * 4) + lane * 16);
            asm volatile("global_load_async_to_lds_b128 %0, %1, off"
                         :: "v"(l), "v"(g) : "memory");
        }
        // loads must land in LDS before the async store engine reads it
        asm volatile("s_wait_asynccnt 0x0" ::: "memory");

        // ---- async stores: column shift + clamp via per-lane LDS address ----
        for (int k = 0; k < 32; ++k) {
            const int p = k * THREADS + t;                 // 0..8191 in half
            const int r = p >> 7;                          // relative row
            const int j = p & (H - 1);                     // output column
            const int xc = clamp127(j + sx);               // column clamp
            const unsigned l = lbase + (unsigned)(r * (H * 4) + xc * 4);
            float* g = dst + (size_t)(i0 + r) * H + j;     // coalesced b32
            asm volatile("global_store_async_from_lds_b32 %0, %1, off"
                         :: "v"(g), "v"(l) : "memory");
        }
        // stores must drain before LDS rows are overwritten next half
        asm volatile("s_wait_asynccnt 0x0" ::: "memory");
    }
}

extern "C" void kernel_launch(void* const* d_in, const int* in_sizes, int n_in,
                              void* d_out, int out_size, void* d_ws, size_t ws_size,
                              hipStream_t stream) {
    const float* x     = (const float*)d_in[0];  // [512, 9, 128, 128] f32
    const int*   shift = (const int*)d_in[1];    // [512, 1, 1, 2] i32 in [0, 8]
    float*       out   = (float*)d_out;          // [512, 9, 128, 128] f32

    const int images = in_sizes[0] / IMG_ELEMS;  // 512 * 9 = 4608
    RandomShiftsAug_kernel<<<dim3(images), dim3(THREADS), 0, stream>>>(x, shift, out);
}